// MemStream_46918222742382
// MI455X (gfx1250) — compile-verified
//
#include <hip/hip_runtime.h>
#include <math.h>

#define IN_DIM   1024
#define OUT_DIM  128
#define MEM_LEN  131072
#define NBLK     2048
#define ROWS_PER_BLK (MEM_LEN / NBLK)   // 64
#define GAMMA    0.5f

typedef float v2f __attribute__((ext_vector_type(2)));
typedef float v8f __attribute__((ext_vector_type(8)));
typedef float f4a __attribute__((ext_vector_type(4)));              // 16B-aligned
typedef float f4u __attribute__((ext_vector_type(4), aligned(4)));  // d_out+1 is 4B-aligned

__device__ __forceinline__ float waveReduceSum(float v) {
#pragma unroll
    for (int off = 16; off > 0; off >>= 1) v += __shfl_xor(v, off, 32);
    return v;
}

// ---------------------------------------------------------------------------
// Kernel 1: normalize, enc = tanh(new@W_enc+b_enc) via V_WMMA_F32_16X16X4_F32,
// rec = enc@W_dec+b_dec via WMMA, mse -> win_loss. One block, 8 waves.
// A-operand: only row M=0 is live; build it branchlessly (ds_load_b64 + mask
// multiply) so the K-loop has no exec-mask save/restore.
// ---------------------------------------------------------------------------
__global__ __launch_bounds__(256)
void encode_kernel(const float* __restrict__ x, const float* __restrict__ mean,
                   const float* __restrict__ stdv,
                   const float* __restrict__ W_enc, const float* __restrict__ b_enc,
                   const float* __restrict__ W_dec, const float* __restrict__ b_dec,
                   const float* __restrict__ win_mean, const float* __restrict__ win_std,
                   float* __restrict__ ws_enc, float* __restrict__ ws_winloss)
{
    __shared__ __align__(16) float s_new[IN_DIM];
    __shared__ __align__(16) float s_enc[OUT_DIM];
    __shared__ float s_red[8];

    const int tid  = threadIdx.x;
    const int lane = tid & 31;
    const int wave = tid >> 5;
    const bool hi  = (lane >= 16);
    const int  l   = lane & 15;
    const float amask = (l == 0) ? 1.0f : 0.0f;   // A row-0 mask, computed once

    for (int i = tid; i < IN_DIM; i += 256) {
        float s = stdv[i];
        s_new[i] = (s == 0.0f) ? 0.0f : (x[i] - mean[i]) / s;
    }
    __syncthreads();

    // ---- encoder: wave w owns output cols [16w, 16w+16).
    {
        const int n0 = wave * 16;
        const v2f* pn = ((const v2f*)s_new) + (hi ? 1 : 0);  // lane's K pair base
        v8f acc = {0.f,0.f,0.f,0.f,0.f,0.f,0.f,0.f};
        for (int k = 0; k < IN_DIM; k += 4) {
            v2f a = pn[k >> 1];                   // ds_load_b64: s_new[k0], s_new[k0+1]
            a.x *= amask; a.y *= amask;
            const int k0 = k + (hi ? 2 : 0);
            v2f b;
            b.x = W_enc[(size_t)k0       * OUT_DIM + n0 + l];
            b.y = W_enc[(size_t)(k0 + 1) * OUT_DIM + n0 + l];
            acc = __builtin_amdgcn_wmma_f32_16x16x4_f32(false, a, false, b,
                                                        (short)0, acc, false, false);
        }
        if (!hi) {  // D row M=0 -> VGPR0, lanes 0..15
            float e = tanhf(acc[0] + b_enc[n0 + l]);
            s_enc[n0 + l]  = e;
            ws_enc[n0 + l] = e;
        }
    }
    __syncthreads();

    // ---- decoder reconstruction + MSE accumulation
    float sq = 0.0f;
    {
        const v2f* pe = ((const v2f*)s_enc) + (hi ? 1 : 0);
        for (int t = wave; t < IN_DIM / 16; t += 8) {
            const int n0 = t * 16;
            v8f acc = {0.f,0.f,0.f,0.f,0.f,0.f,0.f,0.f};
            for (int k = 0; k < OUT_DIM; k += 4) {
                v2f a = pe[k >> 1];
                a.x *= amask; a.y *= amask;
                const int k0 = k + (hi ? 2 : 0);
                v2f b;
                b.x = W_dec[(size_t)k0       * IN_DIM + n0 + l];
                b.y = W_dec[(size_t)(k0 + 1) * IN_DIM + n0 + l];
                acc = __builtin_amdgcn_wmma_f32_16x16x4_f32(false, a, false, b,
                                                            (short)0, acc, false, false);
            }
            if (!hi) {
                float r = acc[0] + b_dec[n0 + l];
                float d = r - s_new[n0 + l];
                sq += d * d;
            }
        }
    }
    sq = waveReduceSum(sq);
    if (lane == 0) s_red[wave] = sq;
    __syncthreads();
    if (tid == 0) {
        float tot = 0.0f;
        for (int w = 0; w < 8; ++w) tot += s_red[w];
        float mse  = tot / (float)IN_DIM;
        float z    = (mse - win_mean[0]) / win_std[0];
        float prob = 0.5f * erfcf(-z * 0.70710678118654752f);  // ndtr(z)
        ws_winloss[0] = (1.0f - prob) * mse;                   // SKIP_THRESHOLD = 1
    }
}

// ---------------------------------------------------------------------------
// Kernel 2: fused streaming copy + L1 distance + block top-3.
// One wave per memory row (32 lanes x float4 = 512B). NT hints: the 1.15GB
// stream must not thrash the 192MB L2.
// ---------------------------------------------------------------------------
__global__ __launch_bounds__(256)
void stream_kernel(const float* __restrict__ memory, const float* __restrict__ mem_data,
                   const float* __restrict__ ws_enc,
                   float* __restrict__ out_mem, float* __restrict__ out_md,
                   float* __restrict__ cand_d, int* __restrict__ cand_i)
{
    __shared__ float s_enc[OUT_DIM];
    __shared__ float s_d[8 * 3];
    __shared__ int   s_i[8 * 3];

    const int tid  = threadIdx.x;
    const int lane = tid & 31;
    const int wave = tid >> 5;
    const int rowBase = blockIdx.x * ROWS_PER_BLK;

    if (tid < OUT_DIM) s_enc[tid] = ws_enc[tid];
    __syncthreads();

    f4a e;
    e.x = s_enc[lane * 4 + 0]; e.y = s_enc[lane * 4 + 1];
    e.z = s_enc[lane * 4 + 2]; e.w = s_enc[lane * 4 + 3];

    float t0 = INFINITY, t1 = INFINITY, t2 = INFINITY;
    int   i0 = 0, i1 = 0, i2 = 0;

    for (int r = rowBase + wave; r < rowBase + ROWS_PER_BLK; r += 8) {
        const f4a* src = (const f4a*)(memory + (size_t)r * OUT_DIM) + lane;
        f4a v = __builtin_nontemporal_load(src);
        f4u q = v;
        __builtin_nontemporal_store(q, (f4u*)(out_mem + (size_t)r * OUT_DIM + lane * 4));
        float d = fabsf(v.x - e.x) + fabsf(v.y - e.y) +
                  fabsf(v.z - e.z) + fabsf(v.w - e.w);
        d = waveReduceSum(d);                       // all 32 lanes get row dist
        if (d < t0)      { t2 = t1; i2 = i1; t1 = t0; i1 = i0; t0 = d; i0 = r; }
        else if (d < t1) { t2 = t1; i2 = i1; t1 = d;  i1 = r; }
        else if (d < t2) { t2 = d;  i2 = r; }
    }
    if (lane == 0) {
        s_d[wave*3+0] = t0; s_i[wave*3+0] = i0;
        s_d[wave*3+1] = t1; s_i[wave*3+1] = i1;
        s_d[wave*3+2] = t2; s_i[wave*3+2] = i2;
    }

    // bulk copy of mem_data rows owned by this block (256KB/block)
    {
        const f4a* src = (const f4a*)(mem_data + (size_t)rowBase * IN_DIM);
        float*     dst = out_md + (size_t)rowBase * IN_DIM;
        const int nvec = ROWS_PER_BLK * IN_DIM / 4;   // 16384
#pragma unroll 4
        for (int v = tid; v < nvec; v += 256) {
            f4a p = __builtin_nontemporal_load(src + v);
            f4u q = p;
            __builtin_nontemporal_store(q, (f4u*)(dst + (size_t)v * 4));
        }
    }
    __syncthreads();
    if (tid == 0) {
        float b0 = INFINITY, b1 = INFINITY, b2 = INFINITY;
        int   j0 = 0, j1 = 0, j2 = 0;
        for (int n = 0; n < 24; ++n) {
            float d = s_d[n]; int r = s_i[n];
            if (d < b0)      { b2 = b1; j2 = j1; b1 = b0; j1 = j0; b0 = d; j0 = r; }
            else if (d < b1) { b2 = b1; j2 = j1; b1 = d;  j1 = r; }
            else if (d < b2) { b2 = d;  j2 = r; }
        }
        cand_d[blockIdx.x*3+0] = b0; cand_i[blockIdx.x*3+0] = j0;
        cand_d[blockIdx.x*3+1] = b1; cand_i[blockIdx.x*3+1] = j1;
        cand_d[blockIdx.x*3+2] = b2; cand_i[blockIdx.x*3+2] = j2;
    }
}

// ---------------------------------------------------------------------------
// Kernel 3: global top-3 merge, losses, conditional FIFO row scatter.
// ---------------------------------------------------------------------------
__global__ __launch_bounds__(256)
void finalize_kernel(const float* __restrict__ x,
                     const float* __restrict__ ws_enc, const float* __restrict__ ws_winloss,
                     const float* __restrict__ cand_d, const int* __restrict__ cand_i,
                     float* __restrict__ out_loss, float* __restrict__ out_mem,
                     float* __restrict__ out_md)
{
    __shared__ float s_d[256 * 3];
    __shared__ int   s_i[256 * 3];
    __shared__ int   s_sel;

    const int tid = threadIdx.x;
    float t0 = INFINITY, t1 = INFINITY, t2 = INFINITY;
    int   i0 = 0, i1 = 0, i2 = 0;

    for (int n = tid; n < NBLK * 3; n += 256) {
        float d = cand_d[n]; int r = cand_i[n];
        if (d < t0)      { t2 = t1; i2 = i1; t1 = t0; i1 = i0; t0 = d; i0 = r; }
        else if (d < t1) { t2 = t1; i2 = i1; t1 = d;  i1 = r; }
        else if (d < t2) { t2 = d;  i2 = r; }
    }
    s_d[tid*3+0] = t0; s_i[tid*3+0] = i0;
    s_d[tid*3+1] = t1; s_i[tid*3+1] = i1;
    s_d[tid*3+2] = t2; s_i[tid*3+2] = i2;
    __syncthreads();

    if (tid == 0) {
        float b0 = INFINITY, b1 = INFINITY, b2 = INFINITY;
        int   j0 = 0, j1 = 0, j2 = 0;
        for (int n = 0; n < 256 * 3; ++n) {
            float d = s_d[n]; int r = s_i[n];
            if (d < b0)      { b2 = b1; j2 = j1; b1 = b0; j1 = j0; b0 = d; j0 = r; }
            else if (d < b1) { b2 = b1; j2 = j1; b1 = d;  j1 = r; }
            else if (d < b2) { b2 = d;  j2 = r; }
        }
        const float wsum = 1.0f + GAMMA + GAMMA * GAMMA;            // 1.75
        float loss_values = (b0 + GAMMA * b1 + GAMMA * GAMMA * b2) / wsum;
        float win_loss = ws_winloss[0];
        out_loss[0] = win_loss + loss_values;
        s_sel = (loss_values <= win_loss) ? j0 : -1;
    }
    __syncthreads();

    const int sel = s_sel;
    if (sel >= 0) {
        if (tid < OUT_DIM) out_mem[(size_t)sel * OUT_DIM + tid] = ws_enc[tid];
        for (int j = tid; j < IN_DIM; j += 256)
            out_md[(size_t)sel * IN_DIM + j] = x[j];
    }
}

extern "C" void kernel_launch(void* const* d_in, const int* in_sizes, int n_in,
                              void* d_out, int out_size, void* d_ws, size_t ws_size,
                              hipStream_t stream)
{
    const float* x      = (const float*)d_in[0];
    const float* mean   = (const float*)d_in[1];
    const float* stdv   = (const float*)d_in[2];
    const float* memory = (const float*)d_in[3];
    const float* mdata  = (const float*)d_in[4];
    const float* W_enc  = (const float*)d_in[5];
    const float* b_enc  = (const float*)d_in[6];
    const float* W_dec  = (const float*)d_in[7];
    const float* b_dec  = (const float*)d_in[8];
    const float* wmean  = (const float*)d_in[9];
    const float* wstd   = (const float*)d_in[10];

    float* out      = (float*)d_out;
    float* out_loss = out;
    float* out_mem  = out + 1;                                   // 4B-aligned only
    float* out_md   = out + 1 + (size_t)MEM_LEN * OUT_DIM;

    float* wsf    = (float*)d_ws;
    float* ws_enc = wsf;                 // 128 floats
    float* ws_wl  = wsf + 128;           // 1 float
    float* cand_d = wsf + 192;           // 3*NBLK floats
    int*   cand_i = (int*)(wsf + 192 + 3 * NBLK);

    encode_kernel<<<1, 256, 0, stream>>>(x, mean, stdv, W_enc, b_enc, W_dec, b_dec,
                                         wmean, wstd, ws_enc, ws_wl);
    stream_kernel<<<NBLK, 256, 0, stream>>>(memory, mdata, ws_enc, out_mem, out_md,
                                            cand_d, cand_i);
    finalize_kernel<<<1, 256, 0, stream>>>(x, ws_enc, ws_wl, cand_d, cand_i,
                                           out_loss, out_mem, out_md);
}